// LuongAttention_1314259993036
// MI455X (gfx1250) — compile-verified
//
#include <hip/hip_runtime.h>
#include <hip/hip_bf16.h>
#include <math.h>

#define BATCH 64
#define SEQ   2048
#define DM    512
#define NU    512
#define WIN   128.0f

typedef float v2f __attribute__((ext_vector_type(2)));
typedef float v8f __attribute__((ext_vector_type(8)));

// -------------------------------------------------------------------------
// Kernel 1: activated = tanh(query @ Wq^T)   [64,512] = [64,512] x [512,512]
// WMMA f32 16x16x4, one wave per 16x16 output tile, K-loop over 512.
// A[m][k] = query[b0+m][k0+k]; B[k][n] = Wq[u0+n][k0+k]  (Wq.T)
// fp32 A 16x4 layout: lanes0-15 {V0:K=0,V1:K=1}, lanes16-31 {V0:K=2,V1:K=3},
// M = lane%16 in both halves. B 4x16 mirrors it with N = lane%16.
// C/D: VGPR i -> M = i (+8 for upper half-lanes), N = lane%16.
// -------------------------------------------------------------------------
__global__ __launch_bounds__(32) void act_gemm_wmma(const float* __restrict__ q,
                                                    const float* __restrict__ Wq,
                                                    float* __restrict__ act) {
    const int u0   = blockIdx.x * 16;     // N tile (units)
    const int b0   = blockIdx.y * 16;     // M tile (batch)
    const int lane = threadIdx.x;
    const int half = lane >> 4;           // 0: K pair {0,1}, 1: K pair {2,3}
    const int mn   = lane & 15;           // M for A, N for B

    const float* aRow = q  + (size_t)(b0 + mn) * DM;
    const float* bRow = Wq + (size_t)(u0 + mn) * DM;

    v8f c = {};
    #pragma unroll 4
    for (int k0 = 0; k0 < DM; k0 += 4) {
        const int ka = k0 + (half << 1);
        v2f a, b;
        a.x = aRow[ka];     a.y = aRow[ka + 1];
        b.x = bRow[ka];     b.y = bRow[ka + 1];
        c = __builtin_amdgcn_wmma_f32_16x16x4_f32(
                /*neg_a=*/false, a, /*neg_b=*/false, b,
                /*c_mod=*/(short)0, c, /*reuse_a=*/false, /*reuse_b=*/false);
    }

    #pragma unroll
    for (int i = 0; i < 8; ++i) {
        const int row = b0 + i + (half << 3);
        act[(size_t)row * NU + u0 + mn] = tanhf(c[i]);
    }
}

// -------------------------------------------------------------------------
// Kernel 2: per-row scalars. sig = sigmoid(act[b,:]·wp), p = L*sig,
// g = exp(-(L-p)^2/128), start/end = trunc(p -/+ 128). One wave per row.
// -------------------------------------------------------------------------
__global__ __launch_bounds__(32) void sig_kernel(const float* __restrict__ act,
                                                 const float* __restrict__ wp,
                                                 const int*   __restrict__ klen,
                                                 float* __restrict__ gArr,
                                                 int*   __restrict__ sArr,
                                                 int*   __restrict__ eArr) {
    const int b    = blockIdx.x;
    const int lane = threadIdx.x;
    const float* ab = act + (size_t)b * NU;

    float s = 0.0f;
    #pragma unroll
    for (int u = lane * 4; u < NU; u += 128) {
        float4 a = *(const float4*)(ab + u);
        float4 w = *(const float4*)(wp + u);
        s += a.x * w.x + a.y * w.y + a.z * w.z + a.w * w.w;
    }
    #pragma unroll
    for (int off = 16; off; off >>= 1) s += __shfl_xor(s, off, 32);

    if (lane == 0) {
        const float sig = 1.0f / (1.0f + expf(-s));
        const float L   = (float)klen[b];
        const float p   = L * sig;
        const float d   = L - p;
        gArr[b] = expf(-(d * d) / WIN);
        sArr[b] = (int)(p - WIN);   // C cast truncates toward zero == jnp.trunc
        eArr[b] = (int)(p + WIN);
    }
}

// -------------------------------------------------------------------------
// Kernel 3: score[b,s] = query[b,:]·keys[b,s,:]. Memory-bound 256MB stream.
// One wave per key row; lane-private float4 query chunks (registers only);
// 4x b128 loads per row; shfl_xor reduction. 8 rows per wave.
// -------------------------------------------------------------------------
__global__ __launch_bounds__(256) void score_kernel(const float* __restrict__ q,
                                                    const float* __restrict__ keys,
                                                    float* __restrict__ score) {
    const int b     = blockIdx.x >> 5;          // 32 blocks per batch row
    const int stile = blockIdx.x & 31;          // 64 s-positions per block
    const int wave  = threadIdx.x >> 5;
    const int lane  = threadIdx.x & 31;

    const float* qb = q + (size_t)b * DM + lane * 4;
    const float4 q0 = *(const float4*)(qb + 0 * 128);
    const float4 q1 = *(const float4*)(qb + 1 * 128);
    const float4 q2 = *(const float4*)(qb + 2 * 128);
    const float4 q3 = *(const float4*)(qb + 3 * 128);

    const int sbase = stile * 64 + wave * 8;
    const float* krow = keys + ((size_t)b * SEQ + sbase) * DM + lane * 4;

    for (int i = 0; i < 8; ++i) {
        const float* kr = krow + (size_t)i * DM;
        // prefetch next row while this one is consumed
        __builtin_prefetch(kr + DM, 0, 0);
        const float4 k0 = *(const float4*)(kr + 0 * 128);
        const float4 k1 = *(const float4*)(kr + 1 * 128);
        const float4 k2 = *(const float4*)(kr + 2 * 128);
        const float4 k3 = *(const float4*)(kr + 3 * 128);

        float p = k0.x * q0.x + k0.y * q0.y + k0.z * q0.z + k0.w * q0.w;
        p += k1.x * q1.x + k1.y * q1.y + k1.z * q1.z + k1.w * q1.w;
        p += k2.x * q2.x + k2.y * q2.y + k2.z * q2.z + k2.w * q2.w;
        p += k3.x * q3.x + k3.y * q3.y + k3.z * q3.z + k3.w * q3.w;

        #pragma unroll
        for (int off = 16; off; off >>= 1) p += __shfl_xor(p, off, 32);
        if (lane == 0) score[(size_t)b * SEQ + sbase + i] = p;
    }
}

// -------------------------------------------------------------------------
// Kernel 4: per-row softmax stats (max, sum of exp). One block per row.
// -------------------------------------------------------------------------
__global__ __launch_bounds__(256) void stats_kernel(const float* __restrict__ score,
                                                    float* __restrict__ rowmax,
                                                    float* __restrict__ rowsum) {
    __shared__ float red[256];
    const int b = blockIdx.x;
    const int t = threadIdx.x;
    const float* sr = score + (size_t)b * SEQ;

    float mx = -INFINITY;
    for (int s = t; s < SEQ; s += 256) mx = fmaxf(mx, sr[s]);
    red[t] = mx; __syncthreads();
    for (int off = 128; off; off >>= 1) {
        if (t < off) red[t] = fmaxf(red[t], red[t + off]);
        __syncthreads();
    }
    mx = red[0]; __syncthreads();

    float sm = 0.0f;
    for (int s = t; s < SEQ; s += 256) sm += expf(sr[s] - mx);
    red[t] = sm; __syncthreads();
    for (int off = 128; off; off >>= 1) {
        if (t < off) red[t] += red[t + off];
        __syncthreads();
    }
    if (t == 0) { rowmax[b] = mx; rowsum[b] = red[0]; }
}

// -------------------------------------------------------------------------
// Kernel 5: windowed context. Stage <=256 weights in LDS, then coalesced
// accumulation: thread t owns dims t and t+256.
// -------------------------------------------------------------------------
__global__ __launch_bounds__(256) void ctx_kernel(const float* __restrict__ keys,
                                                  const float* __restrict__ score,
                                                  const float* __restrict__ rowmax,
                                                  const float* __restrict__ rowsum,
                                                  const float* __restrict__ gArr,
                                                  const int*   __restrict__ sArr,
                                                  const int*   __restrict__ eArr,
                                                  float* __restrict__ ctx) {
    __shared__ float wlds[256];
    const int b = blockIdx.x;
    const int t = threadIdx.x;

    int s_lo = sArr[b]; if (s_lo < 0) s_lo = 0;
    int s_hi = eArr[b]; if (s_hi > SEQ) s_hi = SEQ;
    int count = s_hi - s_lo; if (count < 0) count = 0;   // count <= 256 by construction

    const float mx  = rowmax[b];
    const float inv = gArr[b] / rowsum[b];
    if (t < count)
        wlds[t] = expf(score[(size_t)b * SEQ + s_lo + t] - mx) * inv;
    __syncthreads();

    float a0 = 0.0f, a1 = 0.0f;
    const float* kb = keys + ((size_t)b * SEQ + s_lo) * DM;
    for (int j = 0; j < count; ++j) {
        const float w = wlds[j];                 // LDS broadcast, no conflicts
        a0 += w * kb[(size_t)j * DM + t];
        a1 += w * kb[(size_t)j * DM + t + 256];
    }
    ctx[(size_t)b * DM + t]       = a0;
    ctx[(size_t)b * DM + t + 256] = a1;
}

// -------------------------------------------------------------------------
extern "C" void kernel_launch(void* const* d_in, const int* in_sizes, int n_in,
                              void* d_out, int out_size, void* d_ws, size_t ws_size,
                              hipStream_t stream) {
    const float* query = (const float*)d_in[0];   // [64,512]
    const float* keys  = (const float*)d_in[1];   // [64,2048,512]
    const int*   klen  = (const int*)  d_in[2];   // [64]
    const float* Wq    = (const float*)d_in[3];   // [512,512]
    const float* wp    = (const float*)d_in[4];   // [1,512]

    float* ctx   = (float*)d_out;                 // [64,512]
    float* score = (float*)d_out + BATCH * DM;    // [64,2048]

    // workspace layout
    float* act    = (float*)d_ws;                 // 64*512 floats
    float* rowmax = act + BATCH * NU;             // 64
    float* rowsum = rowmax + BATCH;               // 64
    float* gArr   = rowsum + BATCH;               // 64
    int*   sArr   = (int*)(gArr + BATCH);         // 64
    int*   eArr   = sArr + BATCH;                 // 64

    // 1. activated = tanh(query @ Wq^T)  (WMMA fp32)
    act_gemm_wmma<<<dim3(NU / 16, BATCH / 16), 32, 0, stream>>>(query, Wq, act);
    // 2. per-row alignment scalars
    sig_kernel<<<BATCH, 32, 0, stream>>>(act, wp, klen, gArr, sArr, eArr);
    // 3. score = q . keys  (dominant 256MB stream)
    score_kernel<<<BATCH * (SEQ / 64), 256, 0, stream>>>(query, keys, score);
    // 4. softmax stats
    stats_kernel<<<BATCH, 256, 0, stream>>>(score, rowmax, rowsum);
    // 5. windowed weighted sum
    ctx_kernel<<<BATCH, 256, 0, stream>>>(keys, score, rowmax, rowsum,
                                          gArr, sArr, eArr, ctx);
}